// MaskMemoryBlock_84748294685204
// MI455X (gfx1250) — compile-verified
//
#include <hip/hip_runtime.h>

typedef __attribute__((ext_vector_type(2))) float v2f;
typedef __attribute__((ext_vector_type(8))) float v8f;

#define B_     256
#define M_     2000
#define F_     16384
#define MPC    200        // memories per class
#define THRESH 2e-4f
#define ENTC   2e-4f
#define COSEPS 1e-8f
#define LOGEPS 1e-12f

// workspace layout in floats
#define WS_DOTS 0
#define WS_WT   (B_ * M_)
#define WS_WNT  (2 * B_ * M_)
#define WS_NZ   (3 * B_ * M_)
#define WS_NM   (3 * B_ * M_ + B_)

__device__ __forceinline__ float block_reduce_sum(float v, float* red) {
    int tid = threadIdx.x;
    red[tid] = v; __syncthreads();
    for (int off = 128; off > 0; off >>= 1) {
        if (tid < off) red[tid] += red[tid + off];
        __syncthreads();
    }
    float r = red[0]; __syncthreads();
    return r;
}

__device__ __forceinline__ float block_reduce_max(float v, float* red) {
    int tid = threadIdx.x;
    red[tid] = v; __syncthreads();
    for (int off = 128; off > 0; off >>= 1) {
        if (tid < off) red[tid] = fmaxf(red[tid], red[tid + off]);
        __syncthreads();
    }
    float r = red[0]; __syncthreads();
    return r;
}

// ---------------- row L2 norms: blocks 0..255 -> z rows, 256..2255 -> memory rows
__global__ __launch_bounds__(256) void norms_kernel(const float* __restrict__ z,
                                                    const float* __restrict__ mem,
                                                    float* __restrict__ ws) {
    __shared__ float red[256];
    int row = blockIdx.x;
    const float* src;
    float* dst;
    if (row < B_) { src = z + (size_t)row * F_;          dst = ws + WS_NZ + row; }
    else          { src = mem + (size_t)(row - B_) * F_; dst = ws + WS_NM + (row - B_); }
    float s = 0.f;
    for (int i = threadIdx.x; i < F_; i += 256) { float v = src[i]; s += v * v; }
    float tot = block_reduce_sum(s, red);
    if (threadIdx.x == 0) *dst = sqrtf(tot);
}

// ---------------- GEMM1: dots[b][m] += zf[b] . memory[m], split-K x8, 32x16 tile/wave
__global__ __launch_bounds__(256) void gemm1_kernel(const float* __restrict__ z,
                                                    const float* __restrict__ mem,
                                                    float* __restrict__ dots) {
    int wid  = (blockIdx.x * 256 + threadIdx.x) >> 5;   // 0..7999
    int lane = threadIdx.x & 31;
    int kc = wid & 7;               // 8 K-chunks of 2048
    int cb = (wid >> 3) % 125;      // 125 col tiles of 16
    int rb = (wid >> 3) / 125;      // 8 row tiles of 32
    int r0 = rb * 32, c0 = cb * 16;
    int kbeg = kc * 2048, kend = kbeg + 2048;

    int m  = lane & 15;
    int ks = (lane >> 4) * 2;       // K sub-offset per ISA A/B layout
    const float* a0p = z   + (size_t)(r0 + m) * F_;
    const float* a1p = z   + (size_t)(r0 + 16 + m) * F_;
    const float* bp  = mem + (size_t)(c0 + m) * F_;     // B(k,n)=memory[n][k]

    v8f acc0 = {}; v8f acc1 = {};
    for (int k = kbeg; k < kend; k += 4) {
        int kk = k + ks;
        v2f a0 = *(const v2f*)(a0p + kk);
        v2f a1 = *(const v2f*)(a1p + kk);
        v2f b  = *(const v2f*)(bp + kk);
        acc0 = __builtin_amdgcn_wmma_f32_16x16x4_f32(false, a0, false, b, (short)0, acc0, false, false);
        acc1 = __builtin_amdgcn_wmma_f32_16x16x4_f32(false, a1, false, b, (short)0, acc1, false, false);
    }
    int ncol  = c0 + (lane & 15);
    int rbase = r0 + 8 * (lane >> 4);   // C/D layout: M = v + 8*(lane>=16), N = lane&15
    for (int v = 0; v < 8; ++v) {
        atomicAdd(&dots[(size_t)(rbase + v) * M_ + ncol],      acc0[v]);
        atomicAdd(&dots[(size_t)(rbase + 16 + v) * M_ + ncol], acc1[v]);
    }
}

// ---------------- softmax + class mask + hardshrink + L1 renorm + entropy
__global__ __launch_bounds__(256) void softmax_mask_kernel(const float* __restrict__ dots,
                                                           const int* __restrict__ labels,
                                                           const float* __restrict__ nzv,
                                                           const float* __restrict__ nmv,
                                                           float* __restrict__ wt,
                                                           float* __restrict__ wnt,
                                                           float* __restrict__ ent_out) {
    __shared__ float buf[M_];
    __shared__ float red[256];
    int b = blockIdx.x, tid = threadIdx.x;
    int label = labels[b];
    float nz = nzv[b];

    float lmax = -3.4e38f;
    for (int j = tid; j < M_; j += 256) {
        float l = dots[(size_t)b * M_ + j] / fmaxf(nz * nmv[j], COSEPS);
        buf[j] = l;
        lmax = fmaxf(lmax, l);
    }
    float rmax = block_reduce_max(lmax, red);

    float lsum = 0.f;
    for (int j = tid; j < M_; j += 256) { float e = expf(buf[j] - rmax); buf[j] = e; lsum += e; }
    float inv = 1.f / block_reduce_sum(lsum, red);

    float st = 0.f, snt = 0.f;
    for (int j = tid; j < M_; j += 256) {
        float w = buf[j] * inv;
        bool tgt = (j / MPC) == label;
        if (w > THRESH) { if (tgt) st += w; else snt += w; }
    }
    float sumt  = block_reduce_sum(st, red);
    float sumnt = block_reduce_sum(snt, red);
    float invt = 1.f / sumt, invnt = 1.f / sumnt;

    float et = 0.f, ent = 0.f;
    for (int j = tid; j < M_; j += 256) {
        float w = buf[j] * inv;
        bool tgt  = (j / MPC) == label;
        bool keep = w > THRESH;
        float a = (keep && tgt)  ? w * invt  : 0.f;
        float c = (keep && !tgt) ? w * invnt : 0.f;
        wt[(size_t)b * M_ + j]  = a;
        wnt[(size_t)b * M_ + j] = c;
        et  -= a * logf(a + LOGEPS);
        ent -= c * logf(c + LOGEPS);
    }
    float tet  = block_reduce_sum(et, red);
    float tent = block_reduce_sum(ent, red);
    if (tid == 0) {
        atomicAdd(&ent_out[0], ENTC * tet);
        atomicAdd(&ent_out[1], ENTC * tent);
    }
}

// ---------------- GEMM2: z_hat = w @ memory, 32x32 tile/wave, both outputs
__global__ __launch_bounds__(256) void gemm2_kernel(const float* __restrict__ wt,
                                                    const float* __restrict__ wnt,
                                                    const float* __restrict__ mem,
                                                    float* __restrict__ out) {
    int wid  = (blockIdx.x * 256 + threadIdx.x) >> 5;   // 0..8191
    int lane = threadIdx.x & 31;
    int cb    = wid & 511;          // 512 col tiles of 32 over F
    int rb    = (wid >> 9) & 7;     // 8 row tiles of 32 over B
    int which = wid >> 12;          // 0: target, 1: non-target
    const float* w = which ? wnt : wt;
    float* o = out + (size_t)which * B_ * F_;
    int r0 = rb * 32, c0 = cb * 32;

    int m  = lane & 15;
    int ks = (lane >> 4) * 2;
    const float* a0p = w + (size_t)(r0 + m) * M_;
    const float* a1p = w + (size_t)(r0 + 16 + m) * M_;
    const float* b0p = mem + (c0 + m);          // B(k,n)=memory[k][c0+n]
    const float* b1p = mem + (c0 + 16 + m);

    v8f acc00 = {}, acc01 = {}, acc10 = {}, acc11 = {};
    for (int k = 0; k < M_; k += 4) {
        int kk = k + ks;
        v2f a0 = *(const v2f*)(a0p + kk);
        v2f a1 = *(const v2f*)(a1p + kk);
        v2f b0; b0.x = b0p[(size_t)kk * F_]; b0.y = b0p[(size_t)(kk + 1) * F_];
        v2f b1; b1.x = b1p[(size_t)kk * F_]; b1.y = b1p[(size_t)(kk + 1) * F_];
        acc00 = __builtin_amdgcn_wmma_f32_16x16x4_f32(false, a0, false, b0, (short)0, acc00, false, false);
        acc01 = __builtin_amdgcn_wmma_f32_16x16x4_f32(false, a0, false, b1, (short)0, acc01, false, false);
        acc10 = __builtin_amdgcn_wmma_f32_16x16x4_f32(false, a1, false, b0, (short)0, acc10, false, false);
        acc11 = __builtin_amdgcn_wmma_f32_16x16x4_f32(false, a1, false, b1, (short)0, acc11, false, false);
    }
    int n0    = c0 + (lane & 15);
    int rbase = r0 + 8 * (lane >> 4);
    for (int v = 0; v < 8; ++v) {
        o[(size_t)(rbase + v) * F_ + n0]           = acc00[v];
        o[(size_t)(rbase + v) * F_ + n0 + 16]      = acc01[v];
        o[(size_t)(rbase + 16 + v) * F_ + n0]      = acc10[v];
        o[(size_t)(rbase + 16 + v) * F_ + n0 + 16] = acc11[v];
    }
}

extern "C" void kernel_launch(void* const* d_in, const int* in_sizes, int n_in,
                              void* d_out, int out_size, void* d_ws, size_t ws_size,
                              hipStream_t stream) {
    const float* z      = (const float*)d_in[0];
    const int*   labels = (const int*)d_in[1];
    const float* mem    = (const float*)d_in[2];
    float* out = (float*)d_out;
    float* ws  = (float*)d_ws;

    float* dots = ws + WS_DOTS;
    float* wt   = ws + WS_WT;
    float* wnt  = ws + WS_WNT;
    float* nz   = ws + WS_NZ;
    float* nm   = ws + WS_NM;
    float* ents = out + 2 * (size_t)B_ * F_;   // two scalar outputs at tail

    hipMemsetAsync(dots, 0, (size_t)B_ * M_ * sizeof(float), stream);
    hipMemsetAsync(ents, 0, 2 * sizeof(float), stream);

    norms_kernel<<<B_ + M_, 256, 0, stream>>>(z, mem, ws);
    gemm1_kernel<<<1000, 256, 0, stream>>>(z, mem, dots);           // 8000 waves
    softmax_mask_kernel<<<B_, 256, 0, stream>>>(dots, labels, nz, nm, wt, wnt, ents);
    gemm2_kernel<<<1024, 256, 0, stream>>>(wt, wnt, mem, out);      // 8192 waves
}